// GraphSage_network2_14087492731321
// MI455X (gfx1250) — compile-verified
//
#include <hip/hip_runtime.h>
#include <math.h>

// GraphSage network forward for MI455X (gfx1250, wave32).
// Heavy math (sage2 100000x256x256 and decode 1024x{1024,384}x{384,192} GEMMs)
// runs on V_WMMA_F32_16X16X4_F32 (f32-exact matrix path). Graph gather/scatter
// stages use L2-resident atomics (working sets << 192MB L2).
// Input pointers assume setup_inputs() dict insertion-order flattening:
// x, edge_index, seg, then params leaf-by-leaf (dense={W,b}, bn={gamma,beta,mean,var}).

#define NN 100000
#define EE 1600000
#define GG 1024

typedef float v2f __attribute__((ext_vector_type(2)));
typedef float v8f __attribute__((ext_vector_type(8)));

// ---------------------------------------------------------------- fills
__global__ void k_fill(float* __restrict__ p, long n, float v) {
  long i = (long)blockIdx.x * blockDim.x + threadIdx.x;
  if (i < n) p[i] = v;
}

// ---------------------------------------------------------------- normalize
__global__ void k_norm(const float* __restrict__ x, float* __restrict__ xn) {
  int i = blockIdx.x * blockDim.x + threadIdx.x;
  if (i >= NN * 5) return;
  int f = i % 5;
  const float T[5] = {0.f, 0.f, -200.f, 10000.f, 0.f};
  const float S[5] = {100.f, 100.f, 100.f, 2500.f, 0.25f};
  xn[i] = (x[i] - T[f]) / S[f];
}

// ---------------------------------------------------------------- edge stage
// edge features -> BN -> message MLP (16->2->2, relu) -> segment stat atomics.
// Messages are relu outputs (>=0) so uint-bit atomics implement float min/max.
__global__ void k_edge(const float* __restrict__ xn, const int* __restrict__ ei,
                       const float* __restrict__ bg, const float* __restrict__ bb,
                       const float* __restrict__ bm, const float* __restrict__ bv,
                       const float* __restrict__ w1, const float* __restrict__ b1,
                       const float* __restrict__ w2, const float* __restrict__ b2,
                       float* smin, float* smax, float* ssum, float* ssq, float* deg) {
  int e = blockIdx.x * blockDim.x + threadIdx.x;
  if (e >= EE) return;
  int i = ei[2 * e], j = ei[2 * e + 1];
  float xi[5], xj[5];
#pragma unroll
  for (int f = 0; f < 5; f++) { xi[f] = xn[i * 5 + f]; xj[f] = xn[j * 5 + f]; }
  float d0 = xj[0] - xi[0], d1 = xj[1] - xi[1], d2 = xj[2] - xi[2];
  float dt = xj[3] - xi[3], dq = xj[4] - xi[4];
  float dd = sqrtf(d0 * d0 + d1 * d1 + d2 * d2);
  float invd = dd > 0.f ? 1.f / dd : 0.f;   // divide_no_nan
  float ef[6] = {dt, dq, dd, d0 * invd, d1 * invd, d2 * invd};
  float m16[16];
#pragma unroll
  for (int f = 0; f < 5; f++) { m16[f] = xi[f]; m16[5 + f] = xj[f]; }
#pragma unroll
  for (int f = 0; f < 6; f++)
    m16[10 + f] = (ef[f] - bm[f]) * rsqrtf(bv[f] + 1e-3f) * bg[f] + bb[f];
  float h0 = b1[0], h1v = b1[1];
#pragma unroll
  for (int f = 0; f < 16; f++) { h0 += m16[f] * w1[2 * f]; h1v += m16[f] * w1[2 * f + 1]; }
  h0 = fmaxf(h0, 0.f); h1v = fmaxf(h1v, 0.f);
  float m0 = fmaxf(b2[0] + h0 * w2[0] + h1v * w2[2], 0.f);
  float m1 = fmaxf(b2[1] + h0 * w2[1] + h1v * w2[3], 0.f);
  atomicMin((unsigned int*)&smin[2 * i],     __float_as_uint(m0));
  atomicMin((unsigned int*)&smin[2 * i + 1], __float_as_uint(m1));
  atomicMax((unsigned int*)&smax[2 * i],     __float_as_uint(m0));
  atomicMax((unsigned int*)&smax[2 * i + 1], __float_as_uint(m1));
  atomicAdd(&ssum[2 * i], m0); atomicAdd(&ssum[2 * i + 1], m1);
  atomicAdd(&ssq[2 * i], m0 * m0); atomicAdd(&ssq[2 * i + 1], m1 * m1);
  atomicAdd(&deg[i], 1.0f);
}

// ---------------------------------------------------------------- node update MLP
__global__ void k_nodeupd(const float* __restrict__ smin, const float* __restrict__ smax,
                          const float* __restrict__ ssum, const float* __restrict__ ssq,
                          const float* __restrict__ deg,
                          const float* __restrict__ u1w, const float* __restrict__ u1b,
                          const float* __restrict__ u2w, const float* __restrict__ u2b,
                          float* __restrict__ h4) {
  int n = blockIdx.x * blockDim.x + threadIdx.x;
  if (n >= NN) return;
  float c = fmaxf(deg[n], 1.f);
  float mu0 = ssum[2 * n] / c, mu1 = ssum[2 * n + 1] / c;
  float emb[8] = {smin[2 * n], smin[2 * n + 1], smax[2 * n], smax[2 * n + 1],
                  mu0, mu1, ssq[2 * n] / c - mu0 * mu0, ssq[2 * n + 1] / c - mu1 * mu1};
  float t[4];
#pragma unroll
  for (int k = 0; k < 4; k++) {
    float a = u1b[k];
#pragma unroll
    for (int f = 0; f < 8; f++) a += emb[f] * u1w[4 * f + k];
    t[k] = fmaxf(a, 0.f);
  }
#pragma unroll
  for (int k = 0; k < 4; k++) {
    float a = u2b[k];
#pragma unroll
    for (int f = 0; f < 4; f++) a += t[f] * u2w[4 * f + k];
    h4[4 * n + k] = fmaxf(a, 0.f);
  }
}

// ---------------------------------------------------------------- neighbor sums
__global__ void k_agg4(const int* __restrict__ ei, const float* __restrict__ h4,
                       float* agg) {
  int t = blockIdx.x * blockDim.x + threadIdx.x;
  if (t >= EE * 4) return;
  int e = t >> 2, k = t & 3;
  atomicAdd(&agg[4 * ei[2 * e] + k], h4[4 * ei[2 * e + 1] + k]);
}

__global__ void k_agg128(const int* __restrict__ ei, const float* __restrict__ h1,
                         float* agg) {
  long t = (long)blockIdx.x * blockDim.x + threadIdx.x;
  if (t >= (long)EE * 128) return;
  int e = (int)(t >> 7), f = (int)(t & 127);
  atomicAdd(&agg[128L * ei[2 * e] + f], h1[128L * ei[2 * e + 1] + f]);
}

// ---------------------------------------------------------------- sage1 (wave/node)
__global__ __launch_bounds__(256) void k_sage1(const float* __restrict__ h4,
                                               const float* __restrict__ agg,
                                               const float* __restrict__ deg,
                                               const float* __restrict__ w,
                                               const float* __restrict__ b,
                                               float* __restrict__ h1) {
  int wv = (blockIdx.x * blockDim.x + threadIdx.x) >> 5;  // node (100000 = 12500*8)
  int lane = threadIdx.x & 31;
  float invd = 1.f / fmaxf(deg[wv], 1.f);
  float in8[8];
#pragma unroll
  for (int f = 0; f < 4; f++) { in8[f] = h4[4 * wv + f]; in8[4 + f] = agg[4 * wv + f] * invd; }
  float acc[4]; float ss = 0.f;
#pragma unroll
  for (int q = 0; q < 4; q++) {
    int c = lane + 32 * q;
    float a = b[c];
#pragma unroll
    for (int f = 0; f < 8; f++) a += in8[f] * w[128 * f + c];
    acc[q] = a; ss += a * a;
  }
  for (int m = 16; m >= 1; m >>= 1) ss += __shfl_xor(ss, m, 32);
  float inv = rsqrtf(fmaxf(ss, 1e-12f));
#pragma unroll
  for (int q = 0; q < 4; q++) h1[128L * wv + lane + 32 * q] = fmaxf(acc[q] * inv, 0.f);
}

// ---------------------------------------------------------------- sage2: WMMA GEMM
// 512 threads = 16 waves; block owns one 16-node row tile; wave w owns output
// cols [16w,16w+16). A = [h1 | agg/deg] staged in padded LDS. Fused epilogue:
// bias + row l2-normalize + relu + pooled min/max/sum atomics (h2 never stored).
// A frag (16x4 f32): lane m=lane&15, half=lane>>4: a={A[m][k0+2h], A[m][k0+2h+1]}.
// C frag (16x16 f32): vgpr v -> row v+8h, col lane&15.
__global__ __launch_bounds__(512) void k_sage2(
    const float* __restrict__ h1, const float* __restrict__ agg,
    const float* __restrict__ deg, const float* __restrict__ W,
    const float* __restrict__ B, const int* __restrict__ seg,
    float* pmax, float* pmin, float* psum, int* gcnt) {
  __shared__ float lds[16 * 257];
  int tid = threadIdx.x;
  int wave = tid >> 5, lane = tid & 31;
  int half = lane >> 4, mc = lane & 15;
  int tile = blockIdx.x;                       // 6250 * 16 = 100000 exact
  for (int q = 0; q < 8; q++) {                // stage 16x256 A tile
    int idx = tid + 512 * q;
    int r = idx >> 8, c = idx & 255;
    int node = tile * 16 + r;
    float v = (c < 128) ? h1[128L * node + c]
                        : agg[128L * node + (c - 128)] / fmaxf(deg[node], 1.f);
    lds[r * 257 + c] = v;
  }
  __syncthreads();
  v8f acc = {};
  int n = 16 * wave + mc;
  for (int k0 = 0; k0 < 256; k0 += 4) {
    int kk = k0 + 2 * half;
    v2f a, b;
    a.x = lds[mc * 257 + kk];
    a.y = lds[mc * 257 + kk + 1];
    b.x = W[(long)kk * 256 + n];
    b.y = W[(long)(kk + 1) * 256 + n];
    __builtin_prefetch(&W[(long)(kk + 4) * 256 + n], 0, 3);
    acc = __builtin_amdgcn_wmma_f32_16x16x4_f32(false, a, false, b, (short)0, acc,
                                                false, false);
  }
  float bias = B[n];
  __syncthreads();                             // all waves done reading A
#pragma unroll
  for (int v = 0; v < 8; v++) lds[(v + 8 * half) * 257 + n] = acc[v] + bias;
  __syncthreads();
  int row = wave;                              // wave w reduces row w
  int node = tile * 16 + row;
  float vals[8]; float ss = 0.f;
#pragma unroll
  for (int q = 0; q < 8; q++) {
    float v = lds[row * 257 + lane + 32 * q];
    vals[q] = v; ss += v * v;
  }
  for (int m = 16; m >= 1; m >>= 1) ss += __shfl_xor(ss, m, 32);
  float inv = rsqrtf(fmaxf(ss, 1e-12f));
  int g = seg[node];
#pragma unroll
  for (int q = 0; q < 8; q++) {
    int c = lane + 32 * q;
    float v = fmaxf(vals[q] * inv, 0.f);       // relu => v >= 0, bit-atomics valid
    atomicMax((unsigned int*)&pmax[256L * g + c], __float_as_uint(v));
    atomicMin((unsigned int*)&pmin[256L * g + c], __float_as_uint(v));
    atomicAdd(&psum[256L * g + c], v);
  }
  if (lane == 0) atomicAdd(&gcnt[g], 1);
}

// ---------------------------------------------------------------- pool concat
__global__ void k_pool_concat(const float* __restrict__ pmax, const float* __restrict__ pmin,
                              const float* __restrict__ psum, const int* __restrict__ gcnt,
                              float* __restrict__ g) {
  int t = blockIdx.x * blockDim.x + threadIdx.x;
  if (t >= GG * 256) return;
  int gi = t >> 8, c = t & 255;
  float cnt = fmaxf((float)gcnt[gi], 1.f);
  float s = psum[t];
  g[1024L * gi + c]       = pmax[t];      // max
  g[1024L * gi + 256 + c] = s / cnt;      // mean
  g[1024L * gi + 512 + c] = s;            // sum
  g[1024L * gi + 768 + c] = pmin[t];      // -max(-x) == min
}

// ---------------------------------------------------------------- decode GEMM (WMMA)
// One wave per 16x16 output tile; fused bias + LeakyReLU(0.15) + BatchNorm.
__global__ __launch_bounds__(32) void k_gemm_lrelu_bn(
    const float* __restrict__ A, const float* __restrict__ W, const float* __restrict__ B,
    const float* __restrict__ gm, const float* __restrict__ bt,
    const float* __restrict__ mn, const float* __restrict__ vr,
    float* __restrict__ out, int K, int Nout) {
  int lane = threadIdx.x & 31;
  int half = lane >> 4, mc = lane & 15;
  int nt = blockIdx.x, mt = blockIdx.y;
  int n = nt * 16 + mc;
  int mrow = mt * 16 + mc;
  v8f acc = {};
  for (int k0 = 0; k0 < K; k0 += 4) {
    int kk = k0 + 2 * half;
    v2f a, b;
    a.x = A[(long)mrow * K + kk];
    a.y = A[(long)mrow * K + kk + 1];
    b.x = W[(long)kk * Nout + n];
    b.y = W[(long)(kk + 1) * Nout + n];
    acc = __builtin_amdgcn_wmma_f32_16x16x4_f32(false, a, false, b, (short)0, acc,
                                                false, false);
  }
  float bias = B[n];
  float gmv = gm[n], btv = bt[n], mnv = mn[n], ivr = rsqrtf(vr[n] + 1e-3f);
#pragma unroll
  for (int v = 0; v < 8; v++) {
    int row = mt * 16 + v + 8 * half;
    float x = acc[v] + bias;
    x = x >= 0.f ? x : 0.15f * x;                    // LeakyReLU(0.15)
    out[(long)row * Nout + n] = (x - mnv) * ivr * gmv + btv;
  }
}

// ---------------------------------------------------------------- output heads
struct HeadP { const float *W1, *b1, *W2, *b2, *W3, *b3; };

__global__ __launch_bounds__(64) void k_heads(const float* __restrict__ gfeat,
                                              HeadP h0, HeadP h1, HeadP h2, HeadP sgp,
                                              float* __restrict__ out) {
  __shared__ float gl[192];
  __shared__ float o1[64];
  __shared__ float red[64];
  int gi = blockIdx.x, t = threadIdx.x;
  gl[t] = gfeat[192L * gi + t];
  gl[t + 64] = gfeat[192L * gi + 64 + t];
  gl[t + 128] = gfeat[192L * gi + 128 + t];
  __syncthreads();
  HeadP hp[4] = {h0, h1, h2, sgp};
#pragma unroll
  for (int h = 0; h < 4; h++) {
    const HeadP p = hp[h];
    float a = p.b1[t];
    for (int f = 0; f < 192; f++) a += gl[f] * p.W1[64 * f + t];
    o1[t] = a;
    __syncthreads();
    float c = p.b2[t];
    for (int f = 0; f < 64; f++) c += o1[f] * p.W2[64 * f + t];
    __syncthreads();                 // done reading o1
    red[t] = c * p.W3[t];
    __syncthreads();
    for (int s = 32; s >= 1; s >>= 1) {
      if (t < s) red[t] += red[t + s];
      __syncthreads();
    }
    if (h < 3) {
      out[388L * gi + 129 * h + t] = a;
      out[388L * gi + 129 * h + 64 + t] = c;
      if (t == 0) out[388L * gi + 129 * h + 128] = red[0] + p.b3[0];
    } else {
      if (t == 0) out[388L * gi + 387] = fabsf(red[0] + p.b3[0]) + 1e-5f;
    }
    __syncthreads();
  }
}

// ---------------------------------------------------------------- host launcher
extern "C" void kernel_launch(void* const* d_in, const int* in_sizes, int n_in,
                              void* d_out, int out_size, void* d_ws, size_t ws_size,
                              hipStream_t stream) {
  (void)in_sizes; (void)n_in; (void)out_size; (void)ws_size;
  const float* x  = (const float*)d_in[0];
  const int* ei   = (const int*)d_in[1];
  const int* seg  = (const int*)d_in[2];
  int p = 3;
  auto F = [&](int i) { return (const float*)d_in[i]; };
  const float *beg = F(p++), *beb = F(p++), *bem = F(p++), *bev = F(p++);   // bn_e
  const float *m1w = F(p++), *m1b = F(p++), *m2w = F(p++), *m2b = F(p++);   // msg1,msg2
  const float *u1w = F(p++), *u1b = F(p++), *u2w = F(p++), *u2b = F(p++);   // upd1,upd2
  const float *s1w = F(p++), *s1b = F(p++), *s2w = F(p++), *s2b = F(p++);   // sage1,sage2
  const float *dW[3], *dB[3];
  for (int i = 0; i < 3; i++) { dW[i] = F(p++); dB[i] = F(p++); }           // dec
  const float *bnG[3], *bnB[3], *bnM[3], *bnV[3];
  for (int i = 0; i < 3; i++) { bnG[i] = F(p++); bnB[i] = F(p++); bnM[i] = F(p++); bnV[i] = F(p++); }
  HeadP hd[3];
  for (int i = 0; i < 3; i++) {
    hd[i].W1 = F(p++); hd[i].b1 = F(p++);
    hd[i].W2 = F(p++); hd[i].b2 = F(p++);
    hd[i].W3 = F(p++); hd[i].b3 = F(p++);
  }
  HeadP sgp;
  sgp.W1 = F(p++); sgp.b1 = F(p++); sgp.W2 = F(p++); sgp.b2 = F(p++);
  sgp.W3 = F(p++); sgp.b3 = F(p++);                                         // sigs, sigs_out

  // Workspace layout (floats): zero-init block, +inf block, then uninitialized.
  float* ws = (float*)d_ws;
  long o = 0;
  long SSUM = o; o += 2L * NN;
  long SSQ  = o; o += 2L * NN;
  long SMAX = o; o += 2L * NN;
  long DEG  = o; o += NN;
  long AGG4 = o; o += 4L * NN;
  long AGG1 = o; o += 128L * NN;
  long PMAXP= o; o += 256L * GG;   // pool max (relu'd values >= 0, so 0-init ok)
  long PSUM = o; o += 256L * GG;
  long GCNT = o; o += GG;
  long ZEND = o;
  long SMIN = o; o += 2L * NN;
  long PMIN = o; o += 256L * GG;
  long IEND = o;
  long XN   = o; o += 5L * NN;
  long H4   = o; o += 4L * NN;
  long H1   = o; o += 128L * NN;
  long GBUF = o; o += 1024L * GG;
  long D1   = o; o += 384L * GG;
  long D2   = o; o += 384L * GG;
  long D3   = o; o += 192L * GG;

  auto NB = [](long n) { return (unsigned)((n + 255) / 256); };

  k_fill<<<NB(ZEND), 256, 0, stream>>>(ws, ZEND, 0.f);
  k_fill<<<NB(IEND - ZEND), 256, 0, stream>>>(ws + ZEND, IEND - ZEND, INFINITY);
  k_norm<<<NB(5L * NN), 256, 0, stream>>>(x, ws + XN);
  k_edge<<<NB(EE), 256, 0, stream>>>(ws + XN, ei, beg, beb, bem, bev,
                                     m1w, m1b, m2w, m2b,
                                     ws + SMIN, ws + SMAX, ws + SSUM, ws + SSQ, ws + DEG);
  k_nodeupd<<<NB(NN), 256, 0, stream>>>(ws + SMIN, ws + SMAX, ws + SSUM, ws + SSQ,
                                        ws + DEG, u1w, u1b, u2w, u2b, ws + H4);
  k_agg4<<<NB(4L * EE), 256, 0, stream>>>(ei, ws + H4, ws + AGG4);
  k_sage1<<<NN / 8, 256, 0, stream>>>(ws + H4, ws + AGG4, ws + DEG, s1w, s1b, ws + H1);
  k_agg128<<<NB(128L * EE), 256, 0, stream>>>(ei, ws + H1, ws + AGG1);
  k_sage2<<<NN / 16, 512, 0, stream>>>(ws + H1, ws + AGG1, ws + DEG, s2w, s2b, seg,
                                       ws + PMAXP, ws + PMIN, ws + PSUM,
                                       (int*)(ws + GCNT));
  k_pool_concat<<<NB(256L * GG), 256, 0, stream>>>(ws + PMAXP, ws + PMIN, ws + PSUM,
                                                   (const int*)(ws + GCNT), ws + GBUF);
  k_gemm_lrelu_bn<<<dim3(24, 64), 32, 0, stream>>>(ws + GBUF, dW[0], dB[0],
      bnG[0], bnB[0], bnM[0], bnV[0], ws + D1, 1024, 384);
  k_gemm_lrelu_bn<<<dim3(24, 64), 32, 0, stream>>>(ws + D1, dW[1], dB[1],
      bnG[1], bnB[1], bnM[1], bnV[1], ws + D2, 384, 384);
  k_gemm_lrelu_bn<<<dim3(12, 64), 32, 0, stream>>>(ws + D2, dW[2], dB[2],
      bnG[2], bnB[2], bnM[2], bnV[2], ws + D3, 384, 192);
  k_heads<<<GG, 64, 0, stream>>>(ws + D3, hd[0], hd[1], hd[2], sgp, (float*)d_out);
}